// NVAR_2705829396529
// MI455X (gfx1250) — compile-verified
//
#include <hip/hip_runtime.h>
#include <cstdint>

// NVAR cubic feature expansion for X:(8,4,4096) fp32 -> out:(8,4,3996,231) fp32.
// Store-bandwidth-bound (~118 MB written, ~0.5 MB read). No contraction => no
// WMMA use; CDNA5 path used: GLOBAL_LOAD_ASYNC_TO_LDS_B32 staging + s_wait_asynccnt.

namespace {
constexpr int K_TAPS  = 10;
constexpr int SKIP    = 2;
constexpr int PAD     = (K_TAPS - 1) * SKIP;   // 18
constexpr int TRANS   = 100;
constexpr int NT_IN   = 4096;
constexpr int NT_OUT  = NT_IN - TRANS;         // 3996
constexpr int NM      = 220;                   // C(12,3) cubic monomials
constexpr int NF      = 1 + K_TAPS + NM;       // 231
constexpr int TT      = 512;                   // time tile per block
constexpr int BLOCK   = 256;

struct MonomTbl { uint32_t v[NM]; };
// Matches itertools.combinations_with_replacement(range(10), 3) order exactly
// (lexicographic i <= j <= l).
constexpr MonomTbl make_monom() {
  MonomTbl t{};
  int m = 0;
  for (int i = 0; i < K_TAPS; ++i)
    for (int j = i; j < K_TAPS; ++j)
      for (int l = j; l < K_TAPS; ++l)
        t.v[m++] = (uint32_t)i | ((uint32_t)j << 8) | ((uint32_t)l << 16);
  return t;
}
constexpr MonomTbl MONOM = make_monom();
} // namespace

__global__ __launch_bounds__(BLOCK) void nvar_features_kernel(
    const float* __restrict__ X, float* __restrict__ out) {
  const int br = blockIdx.x;              // series index: b*4 + r
  const int t0 = blockIdx.y * TT;         // output-time tile start (in [0, 3996))
  const int nt = min(TT, NT_OUT - t0);

  // Input window: xs[u] = X[br, t0 + TRANS - PAD + u]; tap i at output t is
  // xs[(t - t0) + 2*i]. Always in range since t >= TRANS=100 >= PAD=18.
  __shared__ float xs[TT + PAD];

  const float* src = X + (size_t)br * NT_IN + (size_t)(t0 + TRANS - PAD);
  for (int k = (int)threadIdx.x; k < nt + PAD; k += BLOCK) {
    const unsigned loff = (unsigned)(uintptr_t)(&xs[k]);   // low 32b of flat = LDS offset
    const float* g = src + k;
    // CDNA5 async global->LDS (ASYNCcnt-tracked, bypasses VGPRs).
    asm volatile("global_load_async_to_lds_b32 %0, %1, off"
                 :: "v"(loff), "v"(g) : "memory");
  }
  asm volatile("s_wait_asynccnt 0" ::: "memory");
  __syncthreads();

  // Thread tid owns output feature f = tid for every t in the tile.
  const int f = (int)threadIdx.x;
  int di = 0, dj = 0, dl = 0;
  if (f >= 1 + K_TAPS && f < NF) {
    const uint32_t p = MONOM.v[f - (1 + K_TAPS)];
    di = (int)(p & 0xFFu) * SKIP;
    dj = (int)((p >> 8) & 0xFFu) * SKIP;
    dl = (int)((p >> 16) & 0xFFu) * SKIP;
  } else if (f >= 1 && f <= K_TAPS) {
    di = (f - 1) * SKIP;
  }

  if (f >= NF) return;  // lanes 231..255 only helped with the async staging

  float* orow = out + ((size_t)br * NT_OUT + (size_t)t0) * NF + f;

  if (f == 0) {
    #pragma unroll 4
    for (int lt = 0; lt < nt; ++lt)
      __builtin_nontemporal_store(1.0f, orow + (size_t)lt * NF);
  } else if (f <= K_TAPS) {
    #pragma unroll 4
    for (int lt = 0; lt < nt; ++lt)
      __builtin_nontemporal_store(xs[lt + di], orow + (size_t)lt * NF);
  } else {
    #pragma unroll 2
    for (int lt = 0; lt < nt; ++lt) {
      const float v = xs[lt + di] * xs[lt + dj] * xs[lt + dl];
      __builtin_nontemporal_store(v, orow + (size_t)lt * NF);
    }
  }
}

extern "C" void kernel_launch(void* const* d_in, const int* in_sizes, int n_in,
                              void* d_out, int out_size, void* d_ws, size_t ws_size,
                              hipStream_t stream) {
  (void)n_in; (void)out_size; (void)d_ws; (void)ws_size;
  const float* X = (const float*)d_in[0];
  float* out = (float*)d_out;

  const int n_series = in_sizes[0] / NT_IN;          // 8*4 = 32
  dim3 grid((unsigned)n_series, (unsigned)((NT_OUT + TT - 1) / TT));  // 32 x 8
  nvar_features_kernel<<<grid, BLOCK, 0, stream>>>(X, out);
}